// DTSG_90082644066743
// MI455X (gfx1250) — compile-verified
//
#include <hip/hip_runtime.h>
#include <math.h>
#include <stdint.h>

// ---------------------------------------------------------------------------
// Problem constants (match the reference)
// ---------------------------------------------------------------------------
#define PL        128
#define PSTRIDE   64
#define WS        16
#define TILE_ELEMS (PL * PL)          // 16384 floats
#define TILE_BYTES (TILE_ELEMS * 4)   // 65536 bytes = 64 KB

typedef __attribute__((ext_vector_type(4))) unsigned int v4u;
typedef __attribute__((ext_vector_type(8))) int          v8i;
typedef __attribute__((ext_vector_type(4))) int          v4i;

// ---------------------------------------------------------------------------
// TDM descriptor packing (CDNA5 ISA ch. 8, D# groups 0..3)
// ---------------------------------------------------------------------------
// Group 0 (128b): count=1 | lds_addr | global_addr[56:0] | type=2
__device__ __forceinline__ v4u tdm_group0(unsigned lds_addr,
                                          unsigned long long gaddr) {
  v4u g;
  g.x = 1u;                                        // count=1, user descriptor
  g.y = lds_addr;                                  // LDS byte address
  g.z = (unsigned)(gaddr & 0xFFFFFFFFull);         // global_addr[31:0]
  g.w = ((unsigned)(gaddr >> 32) & 0x01FFFFFFu)    // global_addr[56:32]
        | (2u << 30);                              // type = 2 ("image")
  return g;
}

// Group 1 (256b): 1-D tensor of 16384 f32 elements; tile covers whole tensor.
//   data_size=2 (4 bytes), workgroup_mask=0 (not in a cluster),
//   tensor_dim0 = tile_dim0 = 16384, tensor_dim1 = 1, tile_dim1/2 = 0,
//   strides = 16384 (benign for 1-D).
__device__ __forceinline__ v8i tdm_group1(bool iterate) {
  v8i g;
  unsigned d0 = (2u << 16) | (iterate ? (1u << 19) : 0u);
  g[0] = (int)d0;
  g[1] = (int)((TILE_ELEMS & 0xFFFFu) << 16);      // tensor_dim0[15:0]
  g[2] = (int)(((unsigned)TILE_ELEMS >> 16)        // tensor_dim0[31:16]
               | (1u << 16));                      // tensor_dim1[15:0] = 1
  g[3] = (int)((TILE_ELEMS & 0xFFFFu) << 16);      // tile_dim0 = 16384
  g[4] = 0;                                        // tile_dim1 = tile_dim2 = 0
  g[5] = (int)TILE_ELEMS;                          // tensor_dim0_stride[31:0]
  g[6] = (int)((TILE_ELEMS & 0xFFFFu) << 16);      // dim0_stride hi=0 | dim1_stride lo
  g[7] = 0;                                        // dim1_stride hi
  return g;
}

// ---------------------------------------------------------------------------
// patches[p, t, :] = { I[p*64 + t], Q[p*64 + t] }
// ---------------------------------------------------------------------------
__global__ void patches_kernel(const float* __restrict__ I,
                               const float* __restrict__ Q,
                               float2* __restrict__ out, int total) {
  int g = blockIdx.x * blockDim.x + threadIdx.x;
  if (g >= total) return;
  int p = g >> 7;            // / 128
  int t = g & (PL - 1);
  int src = p * PSTRIDE + t;
  out[g] = make_float2(I[src], Q[src]);
}

// ---------------------------------------------------------------------------
// Per workgroup:
//   1. TDM-load edge_weights (64 KB) into LDS          (tensor_load_to_lds)
//   2. in-place sigmoid * band-mask in LDS             (VALU)
//   3. TDM-store LDS tile to `cnt` consecutive 64 KB   (tensor_store_from_lds,
//      output slots via descriptor iteration            iterate_enable=1,
//      with lds_addr_increment = 0                      lds_inc=0, g_inc=16384)
// ---------------------------------------------------------------------------
__global__ void adj_broadcast_kernel(const float* __restrict__ ew,
                                     float* __restrict__ adjs,
                                     int P, int chunk) {
  __shared__ float tile[TILE_ELEMS];   // 64 KB of the 320 KB WGP LDS

  long long start = (long long)blockIdx.x * chunk;
  if (start >= P) return;              // uniform per block
  int cnt = (P - start < chunk) ? (int)(P - start) : chunk;   // <= 31 <= 256

  // Flat->LDS aperture rule (ISA 10.2): low 32 bits of a shared pointer are
  // the LDS byte offset.
  unsigned lds_addr = (unsigned)(unsigned long long)(uintptr_t)&tile[0];

  if (threadIdx.x < 32u) {             // wave 0 only issues TDM ops
    v4u g0 = tdm_group0(lds_addr, (unsigned long long)(uintptr_t)ew);
    v8i g1 = tdm_group1(false);
    v4i g2; g2[0] = 1; g2[1] = 1; g2[2] = TILE_ELEMS; g2[3] = 0;
    v4i g3; g3[0] = 0; g3[1] = 0; g3[2] = 0; g3[3] = 0;
    v8i g4 = {0, 0, 0, 0, 0, 0, 0, 0};            // unused VADDR4 group
    __builtin_amdgcn_tensor_load_to_lds(g0, g1, g2, g3, g4, 0);
    __builtin_amdgcn_s_wait_tensorcnt(0);
  }
  __syncthreads();

  // adj[i,j] = sigmoid(ew[i,j]) if 0 < |i-j| <= 16 else 0   (in place)
  #pragma unroll
  for (int k = 0; k < TILE_ELEMS / 256; ++k) {
    int idx = (int)threadIdx.x + k * 256;
    int i = idx >> 7;
    int j = idx & (PL - 1);
    int d = i - j; d = d < 0 ? -d : d;
    float v = tile[idx];
    float s = 1.0f / (1.0f + expf(-v));
    tile[idx] = (d > 0 && d <= WS) ? s : 0.0f;
  }
  __syncthreads();

  if (threadIdx.x < 32u) {
    unsigned long long gaddr =
        (unsigned long long)(uintptr_t)(adjs + start * (long long)TILE_ELEMS);
    v4u g0 = tdm_group0(lds_addr, gaddr);
    v8i g1 = tdm_group1(true);                 // iterate_enable = 1
    v4i g2;
    g2[0] = 1;                                 // tensor_dim2 (unused)
    g2[1] = 0;                                 // lds_addr_increment = 0
    g2[2] = TILE_ELEMS;                        // global_addr_increment (elems)
    g2[3] = (cnt - 1) << 16;                   // iterate_count = cnt-1
    v4i g3; g3[0] = 0; g3[1] = 0; g3[2] = 0; g3[3] = 0;
    v8i g4 = {0, 0, 0, 0, 0, 0, 0, 0};            // unused VADDR4 group
    __builtin_amdgcn_tensor_store_from_lds(g0, g1, g2, g3, g4, 0);
    __builtin_amdgcn_s_wait_tensorcnt(0);
  }
}

// ---------------------------------------------------------------------------
extern "C" void kernel_launch(void* const* d_in, const int* in_sizes, int n_in,
                              void* d_out, int out_size, void* d_ws,
                              size_t ws_size, hipStream_t stream) {
  const float* I  = (const float*)d_in[0];
  const float* Q  = (const float*)d_in[1];
  const float* ew = (const float*)d_in[2];

  int L = in_sizes[0];
  int P = (L - PL) / PSTRIDE + 1;          // 7811 for L = 500000

  float*  out     = (float*)d_out;
  float2* patches = (float2*)out;                       // [P,128,2]
  float*  adjs    = out + (size_t)P * PL * 2;           // [P,128,128]

  int totalP = P * PL;
  patches_kernel<<<(totalP + 255) / 256, 256, 0, stream>>>(I, Q, patches,
                                                           totalP);

  const int NB = 256;                      // spread across up to 256 TDM units
  int chunk = (P + NB - 1) / NB;           // 31 iterations per descriptor
  adj_broadcast_kernel<<<NB, 256, 0, stream>>>(ew, adjs, P, chunk);
}